// SegmentBoundaryLearning_2199023255686
// MI455X (gfx1250) — compile-verified
//
#include <hip/hip_runtime.h>

// SegmentBoundaryLearning collapses to: out[b,t,d] = C0 + sum_{i=0..2, dl=-10..10}
//   W[i][dl+10] * feat0[b, t+dl, d+i-1]   (feat zero-extended in t and d)
// WMMA mapping: per 16(t) x 16(d) tile,  Out = sum_i K_i(16x36 band) x X_i(36x16)
// with K_i[m,s] = W[i][s-m] translation-invariant -> precomputed A-fragment table.
// 27 x v_wmma_f32_16x16x4_f32 per tile, full f32 precision.
// Per block: 8 waves = 8 d-tiles; input slab staged in LDS (zero-padded, no
// predicated loads in the hot loop).

typedef __attribute__((ext_vector_type(2))) float v2f;
typedef __attribute__((ext_vector_type(8))) float v8f;

#define B_ 4
#define T_ 256
#define D_ 1024

#define ROWS 36      // t' rows per block slab: [t0-10, t0+25]
#define COLS 130     // d cols per block slab: [d0b-1, d0b+128]
#define CSTR 38      // column stride in floats: even (b64-aligned), bank-spreading

__global__ __launch_bounds__(256) void sbl_stencil_wmma(
    const float* __restrict__ X,      // (B,T,D)
    const float* __restrict__ wL,     // (5,1,3,3)
    const float* __restrict__ bL,     // (5)
    const float* __restrict__ wR,     // (5,1,3,3)
    const float* __restrict__ bR,     // (5)
    const float* __restrict__ fcw,    // (1,5)
    const float* __restrict__ fcb,    // (1)
    float* __restrict__ out)          // (B,T,D)
{
  __shared__ __align__(16) float Ws[64];          // [0..62] = W[3][21], [63] = C0
  __shared__ __align__(16) float Atab[27 * 64];   // A fragments: [i*9+sc][lane*2+v]
  __shared__ __align__(16) float Xs[COLS * CSTR]; // column-major slab [col][row]

  const int tid = threadIdx.x;

  // ---- block decode: 8 d-blocks(128) x 16 t-tiles x 4 batches = 512 blocks ----
  const int d_blk  = blockIdx.x & 7;
  const int t_tile = (blockIdx.x >> 3) & 15;
  const int b      = blockIdx.x >> 7;
  const int d0b = d_blk * 128;
  const int t0  = t_tile * 16;
  const float* Xb = X + (size_t)b * (T_ * D_);

  // ---- phase 1: fold convs + fc + mean into the 3x21 stencil table ----
  if (tid < 63) {
    const int hh = tid / 21;          // d-offset row of the 3x3 kernels
    const int dl = (tid % 21) - 10;   // time delta
    float acc = 0.f;
    #pragma unroll
    for (int c = 0; c < 5; ++c) {
      const float f = fcw[c];
      const float* wl = wL + c * 9 + hh * 3;
      const float* wr = wR + c * 9 + hh * 3;
      float coef = 0.f;
      if (dl >= -10 && dl <= -3) coef += wl[0];   // left col 0: w2 = 0..7
      if (dl >= -10 && dl <= -2) coef += wl[1];   // left col 1: w2 = 0..8
      if (dl >=  -9 && dl <= -2) coef += wl[2];   // left col 2: w2 = 1..8
      if (dl >=   2 && dl <=  9) coef += wr[0];   // right col 0
      if (dl >=   2 && dl <= 10) coef += wr[1];   // right col 1
      if (dl >=   3 && dl <= 10) coef += wr[2];   // right col 2
      acc += f * coef;
    }
    Ws[tid] = acc * (1.0f / 18.0f);
  } else if (tid == 63) {
    float s = 0.f;
    #pragma unroll
    for (int c = 0; c < 5; ++c) s += fcw[c] * (bL[c] + bR[c]);
    Ws[63] = fcb[0] + 0.5f * s;
  }
  __syncthreads();

  // ---- phase 2a: A-fragment table (same for every tile; 16x4 f32 layout:
  //      M = lane%16, K = 2*(lane/16) + v;  K_i[m,s] = W[i][s-m] banded) ----
  for (int idx = tid; idx < 27 * 64; idx += 256) {
    const int frag   = idx >> 6;          // i*9 + sc
    const int within = idx & 63;          // lane*2 + v
    const int lane_  = within >> 1;
    const int v      = within & 1;
    const int i  = frag / 9;
    const int sc = frag - i * 9;
    const int m  = lane_ & 15;
    const int kh = lane_ >> 4;
    const int e  = sc * 4 + kh * 2 + v - m;
    Atab[idx] = (e >= 0 && e <= 20) ? Ws[i * 21 + e] : 0.f;
  }

  // ---- phase 2b: stage zero-padded slab (clamp + select, no exec branching;
  //      row-major global reads = coalesced; column-major LDS writes) ----
  for (int idx = tid; idx < ROWS * COLS; idx += 256) {
    const int row = idx / COLS;           // t' - (t0-10)
    const int col = idx - row * COLS;     // d  - (d0b-1)
    const int tt = t0 - 10 + row;
    const int dd = d0b - 1 + col;
    const bool inb = ((unsigned)tt < (unsigned)T_) & ((unsigned)dd < (unsigned)D_);
    const int tc = tt < 0 ? 0 : (tt > T_ - 1 ? T_ - 1 : tt);
    const int dc = dd < 0 ? 0 : (dd > D_ - 1 ? D_ - 1 : dd);
    const float val = Xb[(size_t)tc * D_ + dc];
    Xs[col * CSTR + row] = inb ? val : 0.f;
  }
  __syncthreads();

  // ---- phase 3: 27 WMMAs per wave, pure ds_load_b64 operands ----
  const int wave = tid >> 5;
  const int lane = tid & 31;
  const int n  = lane & 15;               // N (and M for the A layout)
  const int kh = lane >> 4;               // K half
  const float C0 = Ws[63];

  v8f acc = {};
  #pragma unroll
  for (int i = 0; i < 3; ++i) {
    const int colbase = wave * 16 + n + i;        // local col of dd = d0w+n+i-1
    #pragma unroll
    for (int sc = 0; sc < 9; ++sc) {
      const int s0 = sc * 4 + kh * 2;             // even -> 8B-aligned b64
      const v2f a  = *(const v2f*)&Atab[(i * 9 + sc) * 64 + lane * 2];
      const v2f bb = *(const v2f*)&Xs[colbase * CSTR + s0];
      acc = __builtin_amdgcn_wmma_f32_16x16x4_f32(
          /*neg_a=*/false, a, /*neg_b=*/false, bb,
          /*c_mod=*/(short)0, acc, /*reuse_a=*/false, /*reuse_b=*/false);
    }
  }

  // ---- store: C/D layout M = r + 8*kh, N = n ----
  float* Ob = out + (size_t)b * (T_ * D_) + (size_t)t0 * D_ + (d0b + wave * 16) + n;
  #pragma unroll
  for (int r = 0; r < 8; ++r) {
    Ob[(size_t)(r + 8 * kh) * D_] = acc[r] + C0;
  }
}

extern "C" void kernel_launch(void* const* d_in, const int* in_sizes, int n_in,
                              void* d_out, int out_size, void* d_ws, size_t ws_size,
                              hipStream_t stream) {
  const float* features     = (const float*)d_in[0];
  const float* conv_left_w  = (const float*)d_in[1];
  const float* conv_left_b  = (const float*)d_in[2];
  const float* conv_right_w = (const float*)d_in[3];
  const float* conv_right_b = (const float*)d_in[4];
  const float* fc_w = (const float*)d_in[5];
  const float* fc_b = (const float*)d_in[6];
  (void)in_sizes; (void)n_in; (void)out_size; (void)d_ws; (void)ws_size;

  sbl_stencil_wmma<<<512, 256, 0, stream>>>(
      features, conv_left_w, conv_left_b, conv_right_w, conv_right_b,
      fc_w, fc_b, (float*)d_out);
}